// Head_88038239634241
// MI455X (gfx1250) — compile-verified
//
#include <hip/hip_runtime.h>
#include <hip/hip_bf16.h>

typedef float v2f __attribute__((ext_vector_type(2)));
typedef float v8f __attribute__((ext_vector_type(8)));

#define BATCH   8
#define TSEQ    2048
#define CDIM    1024
#define HDIM    128
#define MTOT    (BATCH * TSEQ)      // 16384
#define KC      32                  // K-chunk for projection GEMM
#define XS      36                  // padded LDS stride for X tile (floats)
#define KSTR    132                 // padded LDS stride for K tile (floats)
#define PSTR    20                  // padded LDS stride for P staging tile
#define PAIRSTR 288                 // pair-interleaved stride: 256 data + 32 pad words
                                    // (288 mod 64 == 32 -> half-waves hit disjoint banks)

// ---------------------------------------------------------------------------
// Kernel 1: fused QKV projection.  y = x @ W  for W in {Wq, Wk, Wv}.
// Block = 128 threads (4 waves). Block computes 128 rows x 128 cols.
// Each wave: 32 rows (two 16-row tiles) x 128 cols -> 16 v8f accumulators;
// every B-frag (single ds_load_b64 from pair-interleaved W) feeds 2 WMMAs.
// ---------------------------------------------------------------------------
__global__ __launch_bounds__(128) void qkv_proj_kernel(
    const float* __restrict__ X,
    const float* __restrict__ Wk,
    const float* __restrict__ Wq,
    const float* __restrict__ Wv,
    float* __restrict__ qkv)     // [3][MTOT][HDIM]
{
    __shared__ float Xlds[128 * XS];            // 18.0 KB
    __shared__ float Wp[(KC / 2) * PAIRSTR];    // 18.0 KB, pair-interleaved

    const int which = blockIdx.y;                 // 0=q, 1=k, 2=v
    const float* W = (which == 0) ? Wq : (which == 1) ? Wk : Wv;
    float* out = qkv + (size_t)which * MTOT * HDIM;

    const int m0    = blockIdx.x * 128;
    const int tid   = threadIdx.x;
    const int wave  = tid >> 5;
    const int lane  = tid & 31;
    const int lhalf = lane >> 4;                  // 0: K+0/1, 1: K+2/3 (A layout)
    const int ln    = lane & 15;

    v8f acc0[8], acc1[8];
    const v8f vzero = {0.f, 0.f, 0.f, 0.f, 0.f, 0.f, 0.f, 0.f};
#pragma unroll
    for (int t = 0; t < 8; ++t) { acc0[t] = vzero; acc1[t] = vzero; }

    for (int kc = 0; kc < CDIM; kc += KC) {
        // --- stage X tile: rows m0..m0+127, cols kc..kc+31 (one row/thread) ---
        {
            const float* src = X + (size_t)(m0 + tid) * CDIM + kc;
            float* dst = &Xlds[tid * XS];
#pragma unroll
            for (int i = 0; i < 8; ++i)
                *(float4*)(dst + 4 * i) = *(const float4*)(src + 4 * i);
        }
        // --- stage W tile pair-interleaved: Wp[p][c] = {W[kc+2p][c], W[kc+2p+1][c]} ---
        {
            const int p  = tid >> 3;              // 0..15 (k-pair)
            const int c0 = (tid & 7) * 16;        // column start
            const float* sa = W + (size_t)(kc + 2 * p) * HDIM + c0;
            const float* sb = W + (size_t)(kc + 2 * p + 1) * HDIM + c0;
            float4 ra[4], rb[4];
#pragma unroll
            for (int i = 0; i < 4; ++i) { ra[i] = *(const float4*)(sa + 4 * i);
                                          rb[i] = *(const float4*)(sb + 4 * i); }
            float* dst = &Wp[p * PAIRSTR + 2 * c0];
#pragma unroll
            for (int i = 0; i < 4; ++i) {
                v2f w0 = { ra[i].x, rb[i].x }, w1 = { ra[i].y, rb[i].y };
                v2f w2 = { ra[i].z, rb[i].z }, w3 = { ra[i].w, rb[i].w };
                *(v2f*)(dst + 8 * i + 0) = w0;  *(v2f*)(dst + 8 * i + 2) = w1;
                *(v2f*)(dst + 8 * i + 4) = w2;  *(v2f*)(dst + 8 * i + 6) = w3;
            }
        }
        __syncthreads();

#pragma unroll
        for (int j = 0; j < KC / 4; ++j) {
            const int rw = 4 * j + 2 * lhalf;
            // A frags for the two 16-row tiles of this wave
            v2f a0 = *(const v2f*)&Xlds[(wave * 32 + ln) * XS + rw];
            v2f a1 = *(const v2f*)&Xlds[(wave * 32 + 16 + ln) * XS + rw];
            const int prow = (2 * j + lhalf) * PAIRSTR;
#pragma unroll
            for (int t = 0; t < 8; ++t) {
                v2f b = *(const v2f*)&Wp[prow + (t * 16 + ln) * 2];
                acc0[t] = __builtin_amdgcn_wmma_f32_16x16x4_f32(
                    false, a0, false, b, (short)0, acc0[t], false, false);
                acc1[t] = __builtin_amdgcn_wmma_f32_16x16x4_f32(
                    false, a1, false, b, (short)0, acc1[t], false, false);
            }
        }
        __syncthreads();
    }

    // C layout: VGPR r -> row (r | r+8 by half), col = ln
#pragma unroll
    for (int t = 0; t < 8; ++t)
#pragma unroll
        for (int r = 0; r < 8; ++r) {
            const int row = m0 + wave * 32 + (lhalf ? 8 + r : r);
            out[(size_t)row * HDIM + t * 16 + ln]        = acc0[t][r];
            out[(size_t)(row + 16) * HDIM + t * 16 + ln] = acc1[t][r];
        }
}

// ---------------------------------------------------------------------------
// Kernel 2: causal attention with online softmax (flash style).
// Block = 128 threads (4 waves), handles 64 query rows of one batch.
// Each wave: 16 query rows; Q A-frags live in registers (32 x v2f).
// Per 16-key tile: S = Q*K^T (32 wmma), scale*mask*online-softmax,
// P -> LDS -> A-frags, O += P*V (32 wmma, V pair-interleaved in LDS).
// Reference quirk kept: scores multiplied by sqrt(N_EMBD) = 32.
// ---------------------------------------------------------------------------
__global__ __launch_bounds__(128) void attn_kernel(
    const float* __restrict__ qkv, float* __restrict__ out)
{
    __shared__ float Klds[16 * KSTR];          // 8.25 KB
    __shared__ float Vp[8 * PAIRSTR];          // 9.0 KB, pair-interleaved over keys
    __shared__ float Plds[4][16 * PSTR];       // 5.0 KB

    const int b     = blockIdx.x;
    const int q0    = blockIdx.y * 64;
    const int tid   = threadIdx.x;
    const int wave  = tid >> 5;
    const int lane  = tid & 31;
    const int lhalf = lane >> 4;
    const int ln    = lane & 15;

    const float* qg = qkv + ((size_t)b * TSEQ) * HDIM;
    const float* kg = qkv + (size_t)MTOT * HDIM + ((size_t)b * TSEQ) * HDIM;
    const float* vg = qkv + 2 * (size_t)MTOT * HDIM + ((size_t)b * TSEQ) * HDIM;

    const int qrow0 = q0 + wave * 16;
    const float NEG_INF = -__builtin_inff();

    // Preload Q A-frags for all 32 K-steps (K dim = HDIM = 128).
    v2f qa[32];
    {
        const float* qp = qg + (size_t)(qrow0 + ln) * HDIM + 2 * lhalf;
#pragma unroll
        for (int j = 0; j < 32; ++j) qa[j] = *(const v2f*)(qp + 4 * j);
    }

    v8f acc[8];
    const v8f vzero = {0.f, 0.f, 0.f, 0.f, 0.f, 0.f, 0.f, 0.f};
#pragma unroll
    for (int t = 0; t < 8; ++t) acc[t] = vzero;
    float mrow[8], lrow[8];
#pragma unroll
    for (int r = 0; r < 8; ++r) { mrow[r] = NEG_INF; lrow[r] = 0.f; }

    const int nkt = (q0 + 64) / 16;   // causal: keys up to last query in block

    for (int kt = 0; kt < nkt; ++kt) {
        const int k0 = kt * 16;
        // --- stage K tile row-major (B-frags are contiguous b64 along h) ---
        {
            const int row = tid >> 3;             // 0..15
            const int c0  = (tid & 7) * 16;
            const float* ks = kg + (size_t)(k0 + row) * HDIM + c0;
            float* kd = &Klds[row * KSTR + c0];
#pragma unroll
            for (int i = 0; i < 4; ++i)
                *(float4*)(kd + 4 * i) = *(const float4*)(ks + 4 * i);
        }
        // --- stage V pair-interleaved over keys: Vp[p][c] = {V[2p][c], V[2p+1][c]} ---
        {
            const int p  = tid >> 4;              // 0..7 (key pair)
            const int c0 = (tid & 15) * 8;        // column start
            const float* sa = vg + (size_t)(k0 + 2 * p) * HDIM + c0;
            const float* sb = vg + (size_t)(k0 + 2 * p + 1) * HDIM + c0;
            float4 ra[2], rb[2];
#pragma unroll
            for (int i = 0; i < 2; ++i) { ra[i] = *(const float4*)(sa + 4 * i);
                                          rb[i] = *(const float4*)(sb + 4 * i); }
            float* dst = &Vp[p * PAIRSTR + 2 * c0];
#pragma unroll
            for (int i = 0; i < 2; ++i) {
                v2f w0 = { ra[i].x, rb[i].x }, w1 = { ra[i].y, rb[i].y };
                v2f w2 = { ra[i].z, rb[i].z }, w3 = { ra[i].w, rb[i].w };
                *(v2f*)(dst + 8 * i + 0) = w0;  *(v2f*)(dst + 8 * i + 2) = w1;
                *(v2f*)(dst + 8 * i + 4) = w2;  *(v2f*)(dst + 8 * i + 6) = w3;
            }
        }
        __syncthreads();

        // --- S = Q * K^T  (16x16, K=128 -> 32 wmma) ---
        v8f s = vzero;
#pragma unroll
        for (int j = 0; j < 32; ++j) {
            // B frag: lane ln holds K[k0+ln][4j+2*lhalf .. +1]
            v2f bf = *(const v2f*)&Klds[ln * KSTR + 4 * j + 2 * lhalf];
            s = __builtin_amdgcn_wmma_f32_16x16x4_f32(
                false, qa[j], false, bf, (short)0, s, false, false);
        }

        // --- scale (x32, matching reference), causal mask, online softmax ---
        const int kcol = k0 + ln;
#pragma unroll
        for (int r = 0; r < 8; ++r) {
            const int qrow = qrow0 + (lhalf ? 8 + r : r);
            float sv = (kcol <= qrow) ? s[r] * 32.0f : NEG_INF;
            float mx = sv;
            mx = fmaxf(mx, __shfl_xor(mx, 1, 16));
            mx = fmaxf(mx, __shfl_xor(mx, 2, 16));
            mx = fmaxf(mx, __shfl_xor(mx, 4, 16));
            mx = fmaxf(mx, __shfl_xor(mx, 8, 16));
            const float mnew = fmaxf(mrow[r], mx);
            const float corr = __expf(mrow[r] - mnew);
            const float p    = __expf(sv - mnew);
            float ps = p;
            ps += __shfl_xor(ps, 1, 16);
            ps += __shfl_xor(ps, 2, 16);
            ps += __shfl_xor(ps, 4, 16);
            ps += __shfl_xor(ps, 8, 16);
            lrow[r] = lrow[r] * corr + ps;
            mrow[r] = mnew;
            s[r] = p;
#pragma unroll
            for (int t = 0; t < 8; ++t) acc[t][r] *= corr;
        }

        // --- convert P from C-layout to A-layout via per-wave LDS tile ---
        float* pw = Plds[wave];
#pragma unroll
        for (int r = 0; r < 8; ++r)
            pw[(lhalf ? 8 + r : r) * PSTR + ln] = s[r];
        v2f pa[4];
#pragma unroll
        for (int j = 0; j < 4; ++j)
            pa[j] = *(const v2f*)&pw[ln * PSTR + 4 * j + 2 * lhalf];

        // --- O += P * V  (16x128, K=16 -> 8 col-tiles x 4 k-steps) ---
#pragma unroll
        for (int j = 0; j < 4; ++j) {
            const int prow = (2 * j + lhalf) * PAIRSTR;
#pragma unroll
            for (int t = 0; t < 8; ++t) {
                v2f bf = *(const v2f*)&Vp[prow + (t * 16 + ln) * 2];
                acc[t] = __builtin_amdgcn_wmma_f32_16x16x4_f32(
                    false, pa[j], false, bf, (short)0, acc[t], false, false);
            }
        }
        __syncthreads();
    }

    // --- epilogue: normalize by row sums and store ---
#pragma unroll
    for (int r = 0; r < 8; ++r) {
        const float inv = 1.0f / lrow[r];
        const int qrow = qrow0 + (lhalf ? 8 + r : r);
        float* op = out + ((size_t)b * TSEQ + qrow) * HDIM + ln;
#pragma unroll
        for (int t = 0; t < 8; ++t) op[t * 16] = acc[t][r] * inv;
    }
}

// ---------------------------------------------------------------------------
extern "C" void kernel_launch(void* const* d_in, const int* in_sizes, int n_in,
                              void* d_out, int out_size, void* d_ws, size_t ws_size,
                              hipStream_t stream) {
    const float* x  = (const float*)d_in[0];
    const float* Wk = (const float*)d_in[1];
    const float* Wq = (const float*)d_in[2];
    const float* Wv = (const float*)d_in[3];
    float* out = (float*)d_out;
    float* qkv = (float*)d_ws;   // [3][16384][128] f32 = 24 MB scratch

    dim3 g1(MTOT / 128, 3);
    qkv_proj_kernel<<<g1, 128, 0, stream>>>(x, Wk, Wq, Wv, qkv);

    dim3 g2(BATCH, TSEQ / 64);
    attn_kernel<<<g2, 128, 0, stream>>>(qkv, out);
}